// ProbaContrasLoss_38525856645687
// MI455X (gfx1250) — compile-verified
//
#include <hip/hip_runtime.h>
#include <hip/hip_bf16.h>
#include <math.h>

#define B_SZ 4096
#define N_SMP 7
#define D_DIM 512
#define LDA 516   // padded LDS row stride (516 % 64 == 4 -> conflict-free across 16 rows)

typedef __attribute__((ext_vector_type(2))) float v2f;
typedef __attribute__((ext_vector_type(8))) float v8f;

// ---------------------------------------------------------------------------
// Kernel 1: build Qs / Ks rows:  out[i] = N * l2norm(mean)[i] + exp(ls)[i] * sum_n eps[i,n,:]
// grid = 2*B blocks (first B -> Qs from query_*, next B -> Ks from target_*), 256 threads
// ---------------------------------------------------------------------------
__global__ void pcl_prep_kernel(const float* __restrict__ qmean, const float* __restrict__ qls,
                                const float* __restrict__ epsq,
                                const float* __restrict__ tmean, const float* __restrict__ tls,
                                const float* __restrict__ epsk,
                                float* __restrict__ Qs, float* __restrict__ Ks) {
    const int b   = blockIdx.x;
    const bool isQ = (b < B_SZ);
    const int row = isQ ? b : (b - B_SZ);
    const float* mean = isQ ? qmean : tmean;
    const float* ls   = isQ ? qls   : tls;
    const float* eps  = isQ ? epsq  : epsk;
    float* out        = isQ ? Qs    : Ks;

    const int d = threadIdx.x * 2;           // each thread owns 2 of the 512 dims
    float2 m = *(const float2*)&mean[(size_t)row * D_DIM + d];

    // block reduction of sum of squares (8 waves of 32)
    float ss = m.x * m.x + m.y * m.y;
    #pragma unroll
    for (int off = 16; off > 0; off >>= 1) ss += __shfl_down(ss, off, 32);

    __shared__ float part[8];
    __shared__ float s_inv;
    const int wid = threadIdx.x >> 5, lane = threadIdx.x & 31;
    if (lane == 0) part[wid] = ss;
    __syncthreads();
    if (threadIdx.x == 0) {
        float t = 0.f;
        #pragma unroll
        for (int w = 0; w < 8; ++w) t += part[w];
        s_inv = 1.0f / fmaxf(sqrtf(t), 1e-12f);
    }
    __syncthreads();
    const float inv = s_inv;

    float2 l = *(const float2*)&ls[(size_t)row * D_DIM + d];
    float ex = __expf(l.x), ey = __expf(l.y);

    float sx = 0.f, sy = 0.f;
    const float* ep = eps + (size_t)row * N_SMP * D_DIM + d;
    #pragma unroll
    for (int n = 0; n < N_SMP; ++n) {
        float2 v = *(const float2*)&ep[(size_t)n * D_DIM];
        sx += v.x; sy += v.y;
    }

    float2 o;
    o.x = (float)N_SMP * m.x * inv + ex * sx;
    o.y = (float)N_SMP * m.y * inv + ey * sy;
    *(float2*)&out[(size_t)row * D_DIM + d] = o;
}

// ---------------------------------------------------------------------------
// Kernel 2: fused GEMM (Qs @ Ks^T / 49) + row-wise online logsumexp + diagonal.
// grid = 256 blocks (one per 16-row M tile), 256 threads = 8 waves.
// Each wave sweeps 32 of the 256 column tiles with V_WMMA_F32_16X16X4_F32.
// Writes per-block partial sum of (lse_i - logit_ii) -> deterministic (no atomics).
// ---------------------------------------------------------------------------
__global__ void __launch_bounds__(256)
pcl_gemm_lse_kernel(const float* __restrict__ Qs, const float* __restrict__ Ks,
                    float* __restrict__ partials) {
    __shared__ float As[16 * LDA];
    __shared__ float diag[16];
    __shared__ float wmaxs[8][16];
    __shared__ float wsums[8][16];
    __shared__ float red[16];

    const int mbase = blockIdx.x * 16;

    // cooperative A-tile load: 16 rows x 512 floats, float4 granularity
    {
        const int tid = threadIdx.x;
        #pragma unroll
        for (int i = 0; i < 8; ++i) {
            int fi   = tid + i * 256;        // 0..2047 float4 chunks
            int row  = fi >> 7;              // 512/4 = 128 chunks per row
            int col4 = fi & 127;
            float4 v = *(const float4*)&Qs[(size_t)(mbase + row) * D_DIM + col4 * 4];
            *(float4*)&As[row * LDA + col4 * 4] = v;
        }
    }
    __syncthreads();

    const int wid  = threadIdx.x >> 5;
    const int lane = threadIdx.x & 31;
    const int half = lane >> 4;              // 0 or 1
    const int col  = lane & 15;

    // per-lane fragment pointers: A[row=col][k0 + 2*half], B[row=nbase+col][k0 + 2*half]
    const float* arow = &As[col * LDA + 2 * half];
    const float scale = 1.0f / (float)(N_SMP * N_SMP);

    float rmax[8], rsum[8];
    #pragma unroll
    for (int r = 0; r < 8; ++r) { rmax[r] = -INFINITY; rsum[r] = 0.f; }

    for (int t = wid; t < B_SZ / 16; t += 8) {
        const int nbase = t * 16;
        const float* brow = &Ks[(size_t)(nbase + col) * D_DIM + 2 * half];

        v8f acc = {};
        #pragma unroll 4
        for (int k0 = 0; k0 < D_DIM; k0 += 4) {
            float2 af = *(const float2*)&arow[k0];
            float2 bf = *(const float2*)&brow[k0];
            v2f a; a[0] = af.x; a[1] = af.y;
            v2f b; b[0] = bf.x; b[1] = bf.y;
            acc = __builtin_amdgcn_wmma_f32_16x16x4_f32(
                false, a, false, b, (short)0, acc, false, false);
        }

        // capture diagonal logits when this is the block's diagonal column tile
        if (t == (int)blockIdx.x) {
            #pragma unroll
            for (int r = 0; r < 8; ++r) {
                int m = r + 8 * half;
                if (col == m) diag[m] = acc[r] * scale;
            }
        }

        // online logsumexp update (lane holds column slot `col` of rows r+8*half)
        #pragma unroll
        for (int r = 0; r < 8; ++r) {
            float v  = acc[r] * scale;
            float nm = fmaxf(rmax[r], v);
            rsum[r]  = rsum[r] * __expf(rmax[r] - nm) + __expf(v - nm);
            rmax[r]  = nm;
        }
    }

    // merge the 16 column-slot partials per row (masks < 16 stay inside the 16-lane group)
    #pragma unroll
    for (int off = 1; off < 16; off <<= 1) {
        #pragma unroll
        for (int r = 0; r < 8; ++r) {
            float om = __shfl_xor(rmax[r], off, 32);
            float os = __shfl_xor(rsum[r], off, 32);
            float nm = fmaxf(rmax[r], om);
            rsum[r]  = rsum[r] * __expf(rmax[r] - nm) + os * __expf(om - nm);
            rmax[r]  = nm;
        }
    }

    if (col == 0) {   // lanes 0 and 16 carry rows 0..7 / 8..15
        #pragma unroll
        for (int r = 0; r < 8; ++r) {
            wmaxs[wid][8 * half + r] = rmax[r];
            wsums[wid][8 * half + r] = rsum[r];
        }
    }
    __syncthreads();

    // cross-wave merge + per-row (lse - diag), then fixed-order block sum
    if (threadIdx.x < 16) {
        int m = threadIdx.x;
        float M = -INFINITY, S = 0.f;
        #pragma unroll
        for (int w = 0; w < 8; ++w) {
            float om = wmaxs[w][m], os = wsums[w][m];
            float nm = fmaxf(M, om);
            S = S * __expf(M - nm) + os * __expf(om - nm);
            M = nm;
        }
        red[m] = (M + __logf(S)) - diag[m];
    }
    __syncthreads();
    if (threadIdx.x == 0) {
        float t = 0.f;
        #pragma unroll
        for (int m = 0; m < 16; ++m) t += red[m];
        partials[blockIdx.x] = t;
    }
}

// ---------------------------------------------------------------------------
// Kernel 3: deterministic tree reduction of 256 block partials -> loss
// ---------------------------------------------------------------------------
__global__ void pcl_finalize_kernel(const float* __restrict__ partials, float* __restrict__ out) {
    __shared__ float s[256];
    s[threadIdx.x] = partials[threadIdx.x];
    __syncthreads();
    #pragma unroll
    for (int off = 128; off > 0; off >>= 1) {
        if (threadIdx.x < (unsigned)off) s[threadIdx.x] += s[threadIdx.x + off];
        __syncthreads();
    }
    if (threadIdx.x == 0) out[0] = s[0] / (float)B_SZ;
}

// ---------------------------------------------------------------------------
extern "C" void kernel_launch(void* const* d_in, const int* in_sizes, int n_in,
                              void* d_out, int out_size, void* d_ws, size_t ws_size,
                              hipStream_t stream) {
    const float* qmean = (const float*)d_in[0];
    const float* qls   = (const float*)d_in[1];
    const float* tmean = (const float*)d_in[2];
    const float* tls   = (const float*)d_in[3];
    const float* epsq  = (const float*)d_in[4];
    const float* epsk  = (const float*)d_in[5];

    float* Qs       = (float*)d_ws;                       // 4096*512 floats = 8 MB
    float* Ks       = Qs + (size_t)B_SZ * D_DIM;          // 8 MB
    float* partials = Ks + (size_t)B_SZ * D_DIM;          // 256 floats

    pcl_prep_kernel<<<2 * B_SZ, 256, 0, stream>>>(qmean, qls, epsq, tmean, tls, epsk, Qs, Ks);
    pcl_gemm_lse_kernel<<<B_SZ / 16, 256, 0, stream>>>(Qs, Ks, partials);
    pcl_finalize_kernel<<<1, 256, 0, stream>>>(partials, (float*)d_out);
}